// TTM_62947040690620
// MI455X (gfx1250) — compile-verified
//
#include <hip/hip_runtime.h>
#include <math.h>

// ---------------------------------------------------------------------------
// CDNA5 (gfx1250) implementation of the 17-token transformer scan.
// Weight-streaming, HBM-bound (~0.85 GB/call, ~8.3 FLOP/B): all big GEMMs use
// V_WMMA_F32_16X16X4_F32 (wave32) with
//   * b128 weight/activation streaming (8 columns per load, .xy/.zw k-steps)
//   * a depth-4 rotating register pipeline (loads issued 4 chunks ahead)
//   * deterministic split-K (grid.y) + reduce/epilogue kernel, giving
//     2k-4k waves per GEMM to cover the 23.3TB/s x ~600ns = ~14MB
//     latency-bandwidth product.
// Activations live as 32-row zero-padded panels so each wave computes two
// 16x16 accumulator tiles (rows 0..15 / 16..31) with no edge handling.
// ---------------------------------------------------------------------------

typedef float v2f __attribute__((ext_vector_type(2)));
typedef float v4f __attribute__((ext_vector_type(4)));
typedef float v8f __attribute__((ext_vector_type(8)));

#define MDIM   2048
#define MROWS  17      // real rows (cls + 16)
#define MPAD   32      // padded rows
#define HA     16
#define DH     128
#define NQKV   (3 * MDIM)
#define NFC1   (4 * MDIM)

// -------------------------- prologue: segment means ------------------------
__global__ void altx_kernel(const float* __restrict__ x, float* __restrict__ altx) {
    int idx = blockIdx.x * 256 + threadIdx.x;          // 32*2048 total
    int r = idx >> 11, m = idx & (MDIM - 1);
    if (r >= 16) { altx[idx] = 0.0f; return; }
    int c1 = m >> 6, c2 = m & 63;
    const float* p = x + (size_t)c2 * (32 * 160) + c1 * 160 + r * 10;
    float s = 0.0f;
#pragma unroll
    for (int j = 0; j < 10; ++j) s += p[j];
    altx[idx] = s * 0.1f;
}

// savespace = concat(cls, altx @ weight^T) + bias, padded to 32 rows.
__global__ void build_ss_kernel(const float* __restrict__ g0, const float* __restrict__ cls,
                                const float* __restrict__ bias, float* __restrict__ ss) {
    int idx = blockIdx.x * 256 + threadIdx.x;          // 32*2048 total
    int r = idx >> 11, m = idx & (MDIM - 1);
    float v;
    if (r == 0)         v = cls[m] + bias[m];
    else if (r < MROWS) v = g0[(size_t)(r - 1) * MDIM + m] + bias[(size_t)r * MDIM + m];
    else                v = 0.0f;
    ss[idx] = v;
}

// ------------------------------- LayerNorm --------------------------------
__global__ void ln_kernel(const float* __restrict__ x, const float* __restrict__ g,
                          const float* __restrict__ b, float* __restrict__ y, int mreal) {
    int r = blockIdx.x;
    int tid = threadIdx.x, lane = tid & 31, wid = tid >> 5;
    const float* xr = x + (size_t)r * MDIM;
    float* yr = y + (size_t)r * MDIM;
    if (r >= mreal) {
        for (int j = tid; j < MDIM; j += 256) yr[j] = 0.0f;
        return;
    }
    float s = 0.0f, q = 0.0f;
    for (int j = tid; j < MDIM; j += 256) { float v = xr[j]; s += v; q += v * v; }
#pragma unroll
    for (int off = 16; off; off >>= 1) {
        s += __shfl_xor(s, off, 32);
        q += __shfl_xor(q, off, 32);
    }
    __shared__ float s1[8], s2[8], mv[2];
    if (lane == 0) { s1[wid] = s; s2[wid] = q; }
    __syncthreads();
    if (tid == 0) {
        float ts = 0.0f, tq = 0.0f;
#pragma unroll
        for (int w = 0; w < 8; ++w) { ts += s1[w]; tq += s2[w]; }
        float m = ts * (1.0f / MDIM);
        mv[0] = m;
        mv[1] = rsqrtf(tq * (1.0f / MDIM) - m * m + 1e-5f);
    }
    __syncthreads();
    float m = mv[0], inv = mv[1];
    for (int j = tid; j < MDIM; j += 256)
        yr[j] = (xr[j] - m) * inv * g[j] + b[j];
}

// Final LN straight into d_out (only the 17 real rows).
__global__ void ln_out_kernel(const float* __restrict__ x, const float* __restrict__ g,
                              const float* __restrict__ b, float* __restrict__ y) {
    int r = blockIdx.x;
    int tid = threadIdx.x, lane = tid & 31, wid = tid >> 5;
    const float* xr = x + (size_t)r * MDIM;
    float* yr = y + (size_t)r * MDIM;
    float s = 0.0f, q = 0.0f;
    for (int j = tid; j < MDIM; j += 256) { float v = xr[j]; s += v; q += v * v; }
#pragma unroll
    for (int off = 16; off; off >>= 1) {
        s += __shfl_xor(s, off, 32);
        q += __shfl_xor(q, off, 32);
    }
    __shared__ float s1[8], s2[8], mv[2];
    if (lane == 0) { s1[wid] = s; s2[wid] = q; }
    __syncthreads();
    if (tid == 0) {
        float ts = 0.0f, tq = 0.0f;
#pragma unroll
        for (int w = 0; w < 8; ++w) { ts += s1[w]; tq += s2[w]; }
        float m = ts * (1.0f / MDIM);
        mv[0] = m;
        mv[1] = rsqrtf(tq * (1.0f / MDIM) - m * m + 1e-5f);
    }
    __syncthreads();
    float m = mv[0], inv = mv[1];
    for (int j = tid; j < MDIM; j += 256)
        yr[j] = (xr[j] - m) * inv * g[j] + b[j];
}

// ----------------------- scalar attention + cumsum ------------------------
__global__ void attn_kernel(const float* __restrict__ qkv, float* __restrict__ imv) {
    int h = blockIdx.x;
    int d = threadIdx.x;                 // 0..127
    int lane = threadIdx.x & 31, wid = threadIdx.x >> 5;
    __shared__ float red[4];
    const float scale = 0.08838834764831845f;   // 1/sqrt(128)
    float acc = 0.0f;
    int col = h * DH + d;
    for (int i = 0; i < MROWS; ++i) {
        const float* row = qkv + (size_t)i * NQKV;
        float qv = row[col];
        float kv = row[MDIM + col];
        float vv = row[2 * MDIM + col];
        float p = qv * kv;
#pragma unroll
        for (int off = 16; off; off >>= 1) p += __shfl_xor(p, off, 32);
        if (lane == 0) red[wid] = p;
        __syncthreads();
        float rsa = (red[0] + red[1] + red[2] + red[3]) * scale;
        __syncthreads();
        float im = rsa * vv;
        if (i < 16) { acc += im; imv[(size_t)i * MDIM + col] = acc; }
        else        imv[(size_t)16 * MDIM + col] = im;
    }
    for (int r = MROWS; r < MPAD; ++r) imv[(size_t)r * MDIM + col] = 0.0f;
}

// ------------------------------ WMMA GEMM ---------------------------------
// partial[ks][i][n] = sum_{k in slice ks} A[i,k] * W[n,k], i in [0,32).
// grid = (N/64, KSPLIT); 4 waves/block, 16 cols/wave; kchunk = K/KSPLIT.
// Per 8 columns: one b128 per operand. Half-wave 0 carries cols k..k+3,
// half-wave 1 carries k+4..k+7 (contraction order inside a WMMA is free as
// long as A and B agree): step1 = .xy -> {k,k+1,k+4,k+5},
//                          step2 = .zw -> {k+2,k+3,k+6,k+7}.
// Depth-4 rotating buffer: loads run 4 chunks (32 cols) ahead of the WMMAs.
__global__ void gemm_wmma_kernel(const float* __restrict__ A, const float* __restrict__ W,
                                 float* __restrict__ partial, int K, int N, int kchunk) {
    constexpr int DEPTH = 4;
    int wid  = threadIdx.x >> 5;
    int lane = threadIdx.x & 31;
    int l15  = lane & 15;
    int half = lane >> 4;
    int nb   = blockIdx.x * 64 + wid * 16;
    int k0   = blockIdx.y * kchunk;

    const float* Wp  = W + (size_t)(nb + l15) * K + k0 + 4 * half;
    const float* A0p = A + (size_t)l15 * K + k0 + 4 * half;
    const float* A1p = A + (size_t)(16 + l15) * K + k0 + 4 * half;

    v4f bA0[DEPTH], bA1[DEPTH], bB[DEPTH];
#pragma unroll
    for (int d = 0; d < DEPTH; ++d) {
        bA0[d] = *(const v4f*)(A0p + 8 * d);
        bA1[d] = *(const v4f*)(A1p + 8 * d);
        bB[d]  = *(const v4f*)(Wp  + 8 * d);
    }

    v8f c0 = {}; v8f c1 = {};
    int nsteps = kchunk >> 3;                    // 8-column chunks (mult of DEPTH)
    for (int s = 0; s < nsteps; s += DEPTH) {
        __builtin_prefetch(Wp + (size_t)(s + 2 * DEPTH) * 8, 0, 1);
#pragma unroll
        for (int d = 0; d < DEPTH; ++d) {
            int snext = s + d + DEPTH;
            int knext = (snext < nsteps) ? (snext << 3) : 0;   // clamped safe addr
            v4f na0 = *(const v4f*)(A0p + knext);
            v4f na1 = *(const v4f*)(A1p + knext);
            v4f nb  = *(const v4f*)(Wp  + knext);
            c0 = __builtin_amdgcn_wmma_f32_16x16x4_f32(false, bA0[d].xy, false, bB[d].xy,
                                                       (short)0, c0, false, false);
            c1 = __builtin_amdgcn_wmma_f32_16x16x4_f32(false, bA1[d].xy, false, bB[d].xy,
                                                       (short)0, c1, false, false);
            c0 = __builtin_amdgcn_wmma_f32_16x16x4_f32(false, bA0[d].zw, false, bB[d].zw,
                                                       (short)0, c0, false, false);
            c1 = __builtin_amdgcn_wmma_f32_16x16x4_f32(false, bA1[d].zw, false, bB[d].zw,
                                                       (short)0, c1, false, false);
            bA0[d] = na0; bA1[d] = na1; bB[d] = nb;
        }
    }

    // C/D layout: VGPR j -> M = j + 8*half, N = lane&15
    float* pout = partial + (size_t)blockIdx.y * MPAD * N;
    int col = nb + l15;
#pragma unroll
    for (int j = 0; j < 8; ++j) {
        int r0 = j + 8 * half;
        pout[(size_t)r0 * N + col]        = c0[j];
        pout[(size_t)(16 + r0) * N + col] = c1[j];
    }
}

// --------------------- split-K reduce + fused epilogue ---------------------
// grid = (N/256, 32). Deterministic fixed-order sum over the S k-slices.
template <bool BIAS, bool GELU, bool RES>
__global__ void reduce_ep_kernel(const float* __restrict__ part, int S, int N,
                                 const float* __restrict__ bias,
                                 const float* __restrict__ res,
                                 float* __restrict__ out, int mreal) {
    int col = blockIdx.x * 256 + threadIdx.x;
    int row = blockIdx.y;
    size_t o = (size_t)row * N + col;
    float v = 0.0f;
    if (row < mreal) {
        for (int s = 0; s < S; ++s)
            v += part[(size_t)(s * MPAD + row) * N + col];
        if (BIAS) v += bias[col];
        if (GELU) v = 0.5f * v * (1.0f + erff(v * 0.70710678118654752f));
        if (RES)  v += res[o];
    }
    out[o] = v;   // pad rows forced to zero -> clean WMMA A-panels downstream
}

// ------------------------------ launcher ----------------------------------
extern "C" void kernel_launch(void* const* d_in, const int* in_sizes, int n_in,
                              void* d_out, int out_size, void* d_ws, size_t ws_size,
                              hipStream_t stream) {
    const float* x      = (const float*)d_in[0];
    const float* weight = (const float*)d_in[1];
    const float* bias   = (const float*)d_in[2];
    const float* cls    = (const float*)d_in[3];
    const float* Wqkv   = (const float*)d_in[4];
    const float* Wo     = (const float*)d_in[5];
    const float* ln1_g  = (const float*)d_in[6];
    const float* ln1_b  = (const float*)d_in[7];
    const float* ln2_g  = (const float*)d_in[8];
    const float* ln2_b  = (const float*)d_in[9];
    const float* ln3_g  = (const float*)d_in[10];
    const float* ln3_b  = (const float*)d_in[11];
    const float* fc1_w  = (const float*)d_in[12];
    const float* fc1_b  = (const float*)d_in[13];
    const float* fc2_w  = (const float*)d_in[14];
    const float* fc2_b  = (const float*)d_in[15];
    float* out = (float*)d_out;

    // workspace layout (f32)
    float* ws   = (float*)d_ws;
    float* altx = ws;                         // 32*2048
    float* g0   = altx + MPAD * MDIM;         // 32*2048
    float* ss   = g0   + MPAD * MDIM;         // 32*2048
    float* xln  = ss   + MPAD * MDIM;         // 32*2048
    float* h2   = xln  + MPAD * MDIM;         // 32*2048
    float* imv  = h2   + MPAD * MDIM;         // 32*2048
    float* qkv  = imv  + MPAD * MDIM;         // 32*6144
    float* h1   = qkv  + MPAD * NQKV;         // 32*8192
    float* part = h1   + MPAD * NFC1;         // up to 8*32*8192 f32 (8 MB)

    // split-K factors: (N/16)*KSPLIT waves per GEMM (~2k-4k waves each)
    const int KS0 = 16, KSQ = 8, KSO = 16, KS1 = 8, KS2 = 16;

    // --- prologue ---
    altx_kernel<<<(MPAD * MDIM) / 256, 256, 0, stream>>>(x, altx);
    gemm_wmma_kernel<<<dim3(MDIM / 64, KS0), 128, 0, stream>>>(
        altx, weight, part, MDIM, MDIM, MDIM / KS0);
    reduce_ep_kernel<false, false, false><<<dim3(MDIM / 256, MPAD), 256, 0, stream>>>(
        part, KS0, MDIM, nullptr, nullptr, g0, 16);
    build_ss_kernel<<<(MPAD * MDIM) / 256, 256, 0, stream>>>(g0, cls, bias, ss);

    // --- 4 transformer blocks (sequential scan) ---
    for (int t = 0; t < 4; ++t) {
        const float* Wqkv_t = Wqkv + (size_t)t * NQKV * MDIM;
        const float* Wo_t   = Wo   + (size_t)t * MDIM * MDIM;

        ln_kernel<<<MPAD, 256, 0, stream>>>(ss, ln1_g, ln1_b, xln, MROWS);

        gemm_wmma_kernel<<<dim3(NQKV / 64, KSQ), 128, 0, stream>>>(
            xln, Wqkv_t, part, MDIM, NQKV, MDIM / KSQ);
        reduce_ep_kernel<false, false, false><<<dim3(NQKV / 256, MPAD), 256, 0, stream>>>(
            part, KSQ, NQKV, nullptr, nullptr, qkv, MROWS);

        attn_kernel<<<HA, DH, 0, stream>>>(qkv, imv);

        gemm_wmma_kernel<<<dim3(MDIM / 64, KSO), 128, 0, stream>>>(
            imv, Wo_t, part, MDIM, MDIM, MDIM / KSO);
        reduce_ep_kernel<false, false, true><<<dim3(MDIM / 256, MPAD), 256, 0, stream>>>(
            part, KSO, MDIM, nullptr, ss, ss, MROWS);

        ln_kernel<<<MPAD, 256, 0, stream>>>(ss, ln2_g, ln2_b, h2, MROWS);

        gemm_wmma_kernel<<<dim3(NFC1 / 64, KS1), 128, 0, stream>>>(
            h2, fc1_w, part, MDIM, NFC1, MDIM / KS1);
        reduce_ep_kernel<true, true, false><<<dim3(NFC1 / 256, MPAD), 256, 0, stream>>>(
            part, KS1, NFC1, fc1_b, nullptr, h1, MROWS);

        gemm_wmma_kernel<<<dim3(MDIM / 64, KS2), 128, 0, stream>>>(
            h1, fc2_w, part, NFC1, MDIM, NFC1 / KS2);
        reduce_ep_kernel<true, false, true><<<dim3(MDIM / 256, MPAD), 256, 0, stream>>>(
            part, KS2, MDIM, fc2_b, ss, ss, MROWS);
    }

    // --- epilogue ---
    ln_out_kernel<<<MROWS, 256, 0, stream>>>(ss, ln3_g, ln3_b, out);
}